// GraphDeepOne_86019605004434
// MI455X (gfx1250) — compile-verified
//
#include <hip/hip_runtime.h>
#include <math.h>

typedef float v2f __attribute__((ext_vector_type(2)));
typedef float v8f __attribute__((ext_vector_type(8)));
typedef int idx_t;   // JAX default (x64 disabled) lowers jnp.int64 -> int32

#define LRELU01(v) ((v) > 0.0f ? (v) : 0.01f * (v))

// ---------------------------------------------------------------------------
// sign-correct float atomic max (int path for >=0, uint-min path for <0)
// ---------------------------------------------------------------------------
__device__ inline void atomicMaxFloat(float* addr, float value) {
    if (value >= 0.0f) {
        atomicMax((int*)addr, __float_as_int(value));
    } else {
        atomicMin((unsigned int*)addr, __float_as_uint(value));
    }
}

// ---------------------------------------------------------------------------
// Persistent fp32-WMMA GEMM: out[r,:] = x[r,:] @ W(64x64) (+ bias).
// Block = 128 threads = 4 waves; wave w owns columns [16w,16w+16).
// The wave's 16-col slice of W (16 k-steps x v2f = 32 VGPRs) is preloaded
// once; the block then grid-strides over 16-row tiles, so the steady-state
// loop is one b64 A-load per v_wmma_f32_16x16x4_f32.
// Lane layouts (ISA 7.12.2, 32-bit):
//   A 16x4 : M = lane%16, K = v + 2*(lane/16)
//   B 4x16 : N = lane%16, K = v + 2*(lane/16)
//   D 16x16: VGPR r -> M = r + 8*(lane/16), N = lane%16
// __syncthreads() between accumulate and store makes in-place (out==x) safe;
// successive tiles of one block touch disjoint rows.
// ---------------------------------------------------------------------------
__global__ void __launch_bounds__(128)
gemm64_wmma(const float* __restrict__ x, const float* __restrict__ W,
            const float* __restrict__ bias, float* __restrict__ out, int nrows) {
    const int lane = threadIdx.x & 31;
    const int wv   = threadIdx.x >> 5;        // 0..3 -> 16-col tile
    const int half = lane >> 4;               // 0/1
    const int l16  = lane & 15;
    const int cb   = wv * 16;

    // register-resident W slice for this wave
    v2f breg[16];
    #pragma unroll
    for (int s = 0; s < 16; ++s) {
        const int ka = s * 4 + 2 * half;      // K = ka + v
        breg[s].x = W[(size_t)ka       * 64 + cb + l16];
        breg[s].y = W[(size_t)(ka + 1) * 64 + cb + l16];
    }
    const float bv = bias ? bias[cb + l16] : 0.0f;

    const int ntiles = (nrows + 15) >> 4;
    for (int t = blockIdx.x; t < ntiles; t += gridDim.x) {
        const int tile = t * 16;
        int arow = tile + l16;                // A-row held by this lane
        if (arow >= nrows) arow = nrows - 1;  // clamp loads (stores guarded)
        const float* __restrict__ xr = x + (size_t)arow * 64;

        v8f acc = {};
        #pragma unroll
        for (int s = 0; s < 16; ++s) {
            const int ka = s * 4 + 2 * half;
            v2f a;
            a.x = xr[ka];
            a.y = xr[ka + 1];
            acc = __builtin_amdgcn_wmma_f32_16x16x4_f32(
                      false, a, false, breg[s], (short)0, acc, false, false);
        }

        __syncthreads();                      // all reads of this stripe done

        #pragma unroll
        for (int r = 0; r < 8; ++r) {
            const int orow = tile + r + 8 * half;
            if (orow < nrows)
                out[(size_t)orow * 64 + cb + l16] = acc[r] + bv;
        }
    }
}

// ---------------------------------------------------------------------------
// per-node attention scores: hs[n] = h[n,:]·a_s, hd[n] = h[n,:]·a_d
// ---------------------------------------------------------------------------
__global__ void node_dots(const float* __restrict__ h,
                          const float* __restrict__ a_s,
                          const float* __restrict__ a_d,
                          float* __restrict__ hs, float* __restrict__ hd, int N) {
    int n = blockIdx.x * blockDim.x + threadIdx.x;
    if (n >= N) return;
    const float4* hp = (const float4*)(h + (size_t)n * 64);
    const float4* ap = (const float4*)a_s;
    const float4* dp = (const float4*)a_d;
    float ss = 0.0f, sd = 0.0f;
    #pragma unroll
    for (int k = 0; k < 16; ++k) {
        float4 v = hp[k], a = ap[k], d = dp[k];
        ss += v.x * a.x + v.y * a.y + v.z * a.z + v.w * a.w;
        sd += v.x * d.x + v.y * d.y + v.z * d.z + v.w * d.w;
    }
    hs[n] = ss;
    hd[n] = sd;
}

__global__ void fill_f32(float* __restrict__ p, float v, long long n) {
    long long i = blockIdx.x * (long long)blockDim.x + threadIdx.x;
    if (i < n) p[i] = v;
}

// ---------------------------------------------------------------------------
// edge pass 1: el = leaky_relu(hs[src]+hd[dst], 0.2); m[dst] = max(m[dst], el)
// edges [0,E) come from edge_index, edges [E,E+N) are self-loops.
// ---------------------------------------------------------------------------
__global__ void edge_scores(const float* __restrict__ hs, const float* __restrict__ hd,
                            const idx_t* __restrict__ ei, long long E, int N,
                            float* __restrict__ el, float* __restrict__ m) {
    long long e = blockIdx.x * (long long)blockDim.x + threadIdx.x;
    if (e >= E + N) return;
    int sN, dN;
    if (e < E) { sN = ei[e]; dN = ei[E + e]; } else { sN = dN = (int)(e - E); }
    float v = hs[sN] + hd[dN];
    v = v > 0.0f ? v : 0.2f * v;
    el[e] = v;
    atomicMaxFloat(m + dN, v);
}

// edge pass 2: p = exp(el - m[dst]); s[dst] += p  (el overwritten with p)
__global__ void edge_exp_sum(float* __restrict__ el, const float* __restrict__ m,
                             float* __restrict__ s, const idx_t* __restrict__ ei,
                             long long E, int N) {
    long long e = blockIdx.x * (long long)blockDim.x + threadIdx.x;
    if (e >= E + N) return;
    int dN = (e < E) ? ei[E + e] : (int)(e - E);
    float p = expf(el[e] - m[dN]);
    el[e] = p;
    atomicAdd(s + dN, p);
}

// edge pass 3: agg[dst,:] += (p/s[dst]) * h[src,:]   (wave32 per edge, 2 ch/lane)
__global__ void edge_aggregate(const float* __restrict__ p, const float* __restrict__ s,
                               const float* __restrict__ h, const idx_t* __restrict__ ei,
                               long long E, int N, float* __restrict__ agg) {
    long long gid  = blockIdx.x * (long long)blockDim.x + threadIdx.x;
    long long e    = gid >> 5;
    int       lane = (int)(gid & 31);
    if (e >= E + N) return;
    int sN, dN;
    if (e < E) { sN = ei[e]; dN = ei[E + e]; } else { sN = dN = (int)(e - E); }
    float alpha = p[e] / s[dN];
    float2 v = ((const float2*)(h + (size_t)sN * 64))[lane];
    float* dstp = agg + (size_t)dN * 64 + lane * 2;
    atomicAdd(dstp + 0, alpha * v.x);
    atomicAdd(dstp + 1, alpha * v.y);
}

// out[n,c] = leaky_relu(agg[n,c] + b[c], 0.01)
__global__ void finalize_nodes(const float* __restrict__ agg, const float* __restrict__ b,
                               float* __restrict__ out, int N) {
    long long i = blockIdx.x * (long long)blockDim.x + threadIdx.x;
    if (i >= (long long)N * 64) return;
    float v = agg[i] + b[i & 63];
    out[i] = LRELU01(v);
}

// ---------------------------------------------------------------------------
// attentional pooling (per-graph, per-channel softmax over nodes)
// ---------------------------------------------------------------------------
__global__ void pool_max(const float* __restrict__ gate, const idx_t* __restrict__ batch,
                         float* __restrict__ gm, int N) {
    long long i = blockIdx.x * (long long)blockDim.x + threadIdx.x;
    if (i >= (long long)N * 64) return;
    int n = (int)(i >> 6), c = (int)(i & 63);
    atomicMaxFloat(gm + (size_t)batch[n] * 64 + c, gate[i]);
}

__global__ void pool_exp_sum(float* __restrict__ gate, const float* __restrict__ gm,
                             float* __restrict__ gs, const idx_t* __restrict__ batch, int N) {
    long long i = blockIdx.x * (long long)blockDim.x + threadIdx.x;
    if (i >= (long long)N * 64) return;
    int n = (int)(i >> 6), c = (int)(i & 63);
    int g = batch[n];
    float p = expf(gate[i] - gm[(size_t)g * 64 + c]);
    gate[i] = p;
    atomicAdd(gs + (size_t)g * 64 + c, p);
}

__global__ void pool_agg(const float* __restrict__ gate, const float* __restrict__ gs,
                         const float* __restrict__ h, const idx_t* __restrict__ batch,
                         float* __restrict__ pooled, int N) {
    long long i = blockIdx.x * (long long)blockDim.x + threadIdx.x;
    if (i >= (long long)N * 64) return;
    int n = (int)(i >> 6), c = (int)(i & 63);
    int g = batch[n];
    float attn = gate[i] / gs[(size_t)g * 64 + c];
    atomicAdd(pooled + (size_t)g * 64 + c, attn * h[i]);
}

// ---------------------------------------------------------------------------
// tiny MLP layer: out = leaky_relu(in @ W + b, 0.01); thread per (row, col)
// ---------------------------------------------------------------------------
__global__ void linear_leaky(const float* __restrict__ in, const float* __restrict__ W,
                             const float* __restrict__ b, float* __restrict__ out,
                             int rows, int K, int Cout) {
    int idx = blockIdx.x * blockDim.x + threadIdx.x;
    if (idx >= rows * Cout) return;
    int r = idx / Cout, c = idx % Cout;
    float acc = b[c];
    const float* ir = in + (size_t)r * K;
    for (int k = 0; k < K; ++k) acc += ir[k] * W[(size_t)k * Cout + c];
    out[idx] = LRELU01(acc);
}

// ---------------------------------------------------------------------------
extern "C" void kernel_launch(void* const* d_in, const int* in_sizes, int n_in,
                              void* d_out, int out_size, void* d_ws, size_t ws_size,
                              hipStream_t stream) {
    (void)n_in; (void)out_size; (void)ws_size;

    const float* x     = (const float*)d_in[0];
    const idx_t* ei    = (const idx_t*)d_in[1];
    const idx_t* batch = (const idx_t*)d_in[2];
    const float* Wl[3]  = {(const float*)d_in[3],  (const float*)d_in[7],  (const float*)d_in[11]};
    const float* bl[3]  = {(const float*)d_in[4],  (const float*)d_in[8],  (const float*)d_in[12]};
    const float* asl[3] = {(const float*)d_in[5],  (const float*)d_in[9],  (const float*)d_in[13]};
    const float* adl[3] = {(const float*)d_in[6],  (const float*)d_in[10], (const float*)d_in[14]};
    const float* gW  = (const float*)d_in[15];
    const float* gb  = (const float*)d_in[16];
    const float* lW0 = (const float*)d_in[17];
    const float* lb0 = (const float*)d_in[18];
    const float* lW1 = (const float*)d_in[19];
    const float* lb1 = (const float*)d_in[20];
    const float* lW2 = (const float*)d_in[21];
    const float* lb2 = (const float*)d_in[22];

    const int       N  = in_sizes[0] / 64;
    const long long E  = in_sizes[1] / 2;
    const int       G  = 512;
    const long long Ep = E + N;

    // workspace carve-up (floats)
    float* A      = (float*)d_ws;                 // N x 64 (features / h, ping)
    float* B      = A + (size_t)N * 64;           // N x 64 (agg / gate, pong)
    float* hs     = B + (size_t)N * 64;           // N
    float* hd     = hs + N;                       // N
    float* m      = hd + N;                       // N
    float* sArr   = m + N;                        // N
    float* el     = sArr + N;                     // E + N
    float* gm     = el + Ep;                      // G x 64
    float* gs     = gm + (size_t)G * 64;          // G x 64
    float* pooled = gs + (size_t)G * 64;          // G x 64
    float* y1     = pooled + (size_t)G * 64;      // G x 128
    float* y2     = y1 + (size_t)G * 128;         // G x 32

    const int BT = 256;
    const int gN   = (N + BT - 1) / BT;
    const int gNC  = (int)(((long long)N * 64 + BT - 1) / BT);
    const int gEp  = (int)((Ep + BT - 1) / BT);
    const int gAgg = (int)((Ep * 32 + BT - 1) / BT);
    int ntiles = (N + 15) / 16;
    const int gGemm = ntiles < 1536 ? ntiles : 1536;   // persistent grid

    const float* cur = x;
    for (int l = 0; l < 3; ++l) {
        gemm64_wmma<<<gGemm, 128, 0, stream>>>(cur, Wl[l], nullptr, A, N);   // h = x@W
        node_dots<<<gN, BT, 0, stream>>>(A, asl[l], adl[l], hs, hd, N);
        fill_f32<<<gN, BT, 0, stream>>>(m, -3.0e38f, N);
        fill_f32<<<gN, BT, 0, stream>>>(sArr, 0.0f, N);
        edge_scores<<<gEp, BT, 0, stream>>>(hs, hd, ei, E, N, el, m);
        edge_exp_sum<<<gEp, BT, 0, stream>>>(el, m, sArr, ei, E, N);
        fill_f32<<<gNC, BT, 0, stream>>>(B, 0.0f, (long long)N * 64);
        edge_aggregate<<<gAgg, BT, 0, stream>>>(el, sArr, A, ei, E, N, B);
        finalize_nodes<<<gNC, BT, 0, stream>>>(B, bl[l], A, N);              // A = next features
        cur = A;
    }

    // attentional aggregation
    gemm64_wmma<<<gGemm, 128, 0, stream>>>(A, gW, gb, B, N);                 // gate
    fill_f32<<<(G * 64 + BT - 1) / BT, BT, 0, stream>>>(gm, -3.0e38f, (long long)G * 64);
    fill_f32<<<(G * 64 + BT - 1) / BT, BT, 0, stream>>>(gs, 0.0f, (long long)G * 64);
    fill_f32<<<(G * 64 + BT - 1) / BT, BT, 0, stream>>>(pooled, 0.0f, (long long)G * 64);
    pool_max<<<gNC, BT, 0, stream>>>(B, batch, gm, N);
    pool_exp_sum<<<gNC, BT, 0, stream>>>(B, gm, gs, batch, N);
    pool_agg<<<gNC, BT, 0, stream>>>(B, gs, A, batch, pooled, N);

    // MLP head
    linear_leaky<<<(G * 128 + BT - 1) / BT, BT, 0, stream>>>(pooled, lW0, lb0, y1, G, 64, 128);
    linear_leaky<<<(G * 32 + BT - 1) / BT, BT, 0, stream>>>(y1, lW1, lb1, y2, G, 128, 32);
    linear_leaky<<<(G + BT - 1) / BT, BT, 0, stream>>>(y2, lW2, lb2, (float*)d_out, G, 32, 1);
}